// LIFNode_21483426414796
// MI455X (gfx1250) — compile-verified
//
#include <hip/hip_runtime.h>

// LIF forward scan, MI455X (gfx1250).
// Pure HBM-streaming kernel: 1 GiB traffic, ~46 us roofline @ 23.3 TB/s.
// Strategy: per-thread float4 column over T=16, depth-4 software pipeline of
// CDNA5 async global->LDS B128 loads (ASYNCcnt), ds_load_b128 consume,
// non-temporal B128 spike stores.

typedef float v4f __attribute__((ext_vector_type(4)));

#define B_    32
#define T_    16
#define S_    (256 * 1024)     // spatial elements per (b, t) slab
#define SV_   (S_ / 4)         // float4 vectors per (b, t) slab = 65536
#define BLK   256              // 8 wave32s per block
#define DEPTH 4                // async pipeline depth (4 x 4KB LDS = 16KB/block)

template <int I> struct IC { static constexpr int v = I; };

template <int I, int N, class F>
__device__ __forceinline__ void static_for(F&& f) {
    if constexpr (I < N) {
        f(IC<I>{});
        static_for<I + 1, N>(f);
    }
}

// Issue an async DMA of 16 bytes/lane from global to LDS (VGLOBAL GV form,
// 64-bit vaddr pair, LDS byte address in a VGPR). Tracked by ASYNCcnt.
// WAIT_DS: when re-targeting an LDS slot we just ds_load'ed from, make sure
// the DS read has completed before the async engine may overwrite it.
template <bool WAIT_DS>
__device__ __forceinline__ void async_load_b128(const v4f* gptr, v4f* lptr) {
    unsigned lds           = (unsigned)(unsigned long long)lptr;  // LDS byte addr
    unsigned long long ga  = (unsigned long long)gptr;
    if constexpr (WAIT_DS) {
        asm volatile("s_wait_dscnt 0\n\t"
                     "global_load_async_to_lds_b128 %0, %1, off"
                     :: "v"(lds), "v"(ga) : "memory");
    } else {
        asm volatile("global_load_async_to_lds_b128 %0, %1, off"
                     :: "v"(lds), "v"(ga) : "memory");
    }
}

template <int N>
__device__ __forceinline__ void wait_asynccnt() {
#if __has_builtin(__builtin_amdgcn_s_wait_asynccnt)
    __builtin_amdgcn_s_wait_asynccnt(N);
#else
    asm volatile("s_wait_asynccnt %0" :: "n"(N) : "memory");
#endif
}

__global__ __launch_bounds__(BLK) void lif_fwd_kernel(const v4f* __restrict__ X,
                                                      v4f* __restrict__ Out) {
    __shared__ v4f sbuf[DEPTH][BLK];   // 16 KB: 4-deep double-buffer per block

    const unsigned gid = blockIdx.x * BLK + threadIdx.x;
    const unsigned b   = gid / SV_;
    const unsigned sv  = gid - b * SV_;
    // float4 index of (b, t=0, sv); successive t are SV_ vectors apart (1 MiB).
    const unsigned long long base = (unsigned long long)b * T_ * SV_ + sv;

    // Prologue: fill the pipeline with the first DEPTH timesteps.
    static_for<0, DEPTH>([&](auto tc) {
        constexpr int t = decltype(tc)::v;
        async_load_b128<false>(X + base + (unsigned long long)t * SV_,
                               &sbuf[t][threadIdx.x]);
    });

    v4f mem = {0.0f, 0.0f, 0.0f, 0.0f};

    static_for<0, T_>([&](auto tc) {
        constexpr int t   = decltype(tc)::v;
        constexpr int rem = T_ - 1 - t;                       // loads issued after load t
        constexpr int wn  = rem < (DEPTH - 1) ? rem : (DEPTH - 1);
        wait_asynccnt<wn>();                                  // load t has landed in LDS

        v4f x = sbuf[t % DEPTH][threadIdx.x];                 // ds_load_b128

        // LIF step: mem += (x + v_reset - mem)/tau, tau=0.5, v_reset=0
        //   -> mem = 2*x - mem; spike = mem > v_th(=1); hard reset.
        v4f spike;
#pragma unroll
        for (int i = 0; i < 4; ++i) {
            float m  = 2.0f * x[i] - mem[i];
            float s  = (m > 1.0f) ? 1.0f : 0.0f;
            spike[i] = s;
            mem[i]   = (s != 0.0f) ? 0.0f : m;
        }

        // Spikes are written once, never re-read: keep them out of L2 (NT).
        __builtin_nontemporal_store(spike, Out + base + (unsigned long long)t * SV_);

        // Refill the slot we just consumed with timestep t+DEPTH.
        if constexpr (t + DEPTH < T_) {
            async_load_b128<true>(X + base + (unsigned long long)(t + DEPTH) * SV_,
                                  &sbuf[t % DEPTH][threadIdx.x]);
        }
    });
}

extern "C" void kernel_launch(void* const* d_in, const int* in_sizes, int n_in,
                              void* d_out, int out_size, void* d_ws, size_t ws_size,
                              hipStream_t stream) {
    (void)in_sizes; (void)n_in; (void)d_ws; (void)ws_size; (void)out_size;
    const v4f* X  = (const v4f*)d_in[0];
    v4f* Out      = (v4f*)d_out;
    // 32 * 65536 float4-threads / 256 = 8192 blocks; every lane active.
    dim3 grid((B_ * SV_) / BLK);
    lif_fwd_kernel<<<grid, dim3(BLK), 0, stream>>>(X, Out);
}